// GCNLayerWithEdge_19636590477405
// MI455X (gfx1250) — compile-verified
//
#include <hip/hip_runtime.h>

#define D 64

typedef __attribute__((ext_vector_type(2))) float v2f;
typedef __attribute__((ext_vector_type(8))) float v8f;

// ---------------------------------------------------------------------------
// Kernel 1: fused  m = node[src]+edge ; online edge-softmax over sorted dst ;
// agg[v] = sum_e m_e * exp(m_e - max)/sum exp   — single pass over each node's
// contiguous edge range. One wave (32 lanes) per destination node, 2 feature
// components per lane (D=64).
// ---------------------------------------------------------------------------
__global__ void __launch_bounds__(256)
gcn_agg_kernel(const float* __restrict__ node_feats,
               const float* __restrict__ edge_feats,
               const int*   __restrict__ src,
               const int*   __restrict__ dst,
               float* __restrict__ agg,
               int N, int E)
{
    const int wave = (blockIdx.x * blockDim.x + threadIdx.x) >> 5;
    const int lane = threadIdx.x & 31;
    if (wave >= N) return;           // wave-uniform exit
    const int v = wave;

    // Binary search the contiguous edge range [start,end) with dst == v.
    // Uniform across the wave -> scalarizes; dst tree levels stay L2-hot.
    int lo = 0, hi = E;
    while (lo < hi) { int mid = (lo + hi) >> 1; if (dst[mid] < v) lo = mid + 1; else hi = mid; }
    const int start = lo;
    hi = E;
    while (lo < hi) { int mid = (lo + hi) >> 1; if (dst[mid] < v + 1) lo = mid + 1; else hi = mid; }
    const int end = lo;

    const float NEG_INF = -__builtin_inff();
    float mx0 = NEG_INF, mx1 = NEG_INF;   // running max per component
    float s0 = 0.f, s1 = 0.f;             // running sum of exp
    float a0 = 0.f, a1 = 0.f;             // running sum of m*exp

    const int f = lane * 2;               // this lane's feature pair
    for (int e = start; e < end; ++e) {
        const int s = src[e];             // wave-uniform -> s_load
        v2f nf = *(const v2f*)(node_feats + s * D + f);
        v2f ef = *(const v2f*)(edge_feats + e * D + f);
        float m0 = nf.x + ef.x;
        float m1 = nf.y + ef.y;

        if (m0 > mx0) { float sc = __expf(mx0 - m0); s0 *= sc; a0 *= sc; mx0 = m0; }
        float e0 = __expf(m0 - mx0);
        s0 += e0; a0 = fmaf(m0, e0, a0);

        if (m1 > mx1) { float sc = __expf(mx1 - m1); s1 *= sc; a1 *= sc; mx1 = m1; }
        float e1 = __expf(m1 - mx1);
        s1 += e1; a1 = fmaf(m1, e1, a1);
    }

    v2f r;
    r.x = a0 / fmaxf(s0, 1e-38f);         // empty node: 0/1e-38 = 0
    r.y = a1 / fmaxf(s1, 1e-38f);
    *(v2f*)(agg + v * D + f) = r;
}

// ---------------------------------------------------------------------------
// Kernel 2: out = relu(agg @ W^T + b) + node_feats  using V_WMMA_F32_16X16X4_F32.
// 256 threads = 8 waves; block stages W transposed in LDS; each wave computes a
// 16-row x 64-col slab as 4 column tiles, accumulating 16 k-steps of 16x16x4.
// Fragment layouts (CDNA5 ISA 7.12.2, 32-bit):
//   A 16x4 : lane l, vgpr v -> A[l%16][2*(l/16)+v]
//   B 4x16 : lane l, vgpr v -> B[2*(l/16)+v][l%16]
//   C/D    : lane l, vgpr r -> D[r+8*(l/16)][l%16]
// ---------------------------------------------------------------------------
__global__ void __launch_bounds__(256)
gcn_gemm_kernel(const float* __restrict__ agg,
                const float* __restrict__ W,      // [D][D], row j = output col j
                const float* __restrict__ bias,   // [D]
                const float* __restrict__ node_feats,
                float* __restrict__ out,
                int N)
{
    __shared__ float Wt[D * D];               // Wt[k][j] = W[j][k]  (B = W^T)
    const int t = threadIdx.x;
    for (int i = t; i < D * D; i += 256) {
        int j = i >> 6, k = i & 63;
        Wt[k * D + j] = W[i];
    }
    __syncthreads();

    const int wave  = t >> 5;
    const int lane  = t & 31;
    const int tile  = blockIdx.x * 8 + wave;  // 16-row tile index
    const int row0  = tile * 16;
    if (row0 >= N) return;                    // wave-uniform; EXEC stays all-1s

    const int mrow  = lane & 15;              // M (A) / N (B,C) index within tile
    const int khalf = lane >> 4;              // which K pair this lane holds

    v8f acc0 = {}, acc1 = {}, acc2 = {}, acc3 = {};

    int arow = row0 + mrow; if (arow >= N) arow = N - 1;   // safety clamp
    const float* aptr = agg + arow * D;

    #pragma unroll
    for (int kb = 0; kb < 16; ++kb) {
        const int k0 = kb * 4 + khalf * 2;
        v2f afrag = *(const v2f*)(aptr + k0);

        v2f b0, b1, b2, b3;
        b0.x = Wt[(k0    ) * D +  0 + mrow]; b0.y = Wt[(k0 + 1) * D +  0 + mrow];
        b1.x = Wt[(k0    ) * D + 16 + mrow]; b1.y = Wt[(k0 + 1) * D + 16 + mrow];
        b2.x = Wt[(k0    ) * D + 32 + mrow]; b2.y = Wt[(k0 + 1) * D + 32 + mrow];
        b3.x = Wt[(k0    ) * D + 48 + mrow]; b3.y = Wt[(k0 + 1) * D + 48 + mrow];

        acc0 = __builtin_amdgcn_wmma_f32_16x16x4_f32(false, afrag, false, b0, (short)0, acc0, false, false);
        acc1 = __builtin_amdgcn_wmma_f32_16x16x4_f32(false, afrag, false, b1, (short)0, acc1, false, false);
        acc2 = __builtin_amdgcn_wmma_f32_16x16x4_f32(false, afrag, false, b2, (short)0, acc2, false, false);
        acc3 = __builtin_amdgcn_wmma_f32_16x16x4_f32(false, afrag, false, b3, (short)0, acc3, false, false);
    }

    // Epilogue: bias + relu + residual, per C/D layout.
    v8f accs[4] = {acc0, acc1, acc2, acc3};
    #pragma unroll
    for (int jt = 0; jt < 4; ++jt) {
        const int col = jt * 16 + mrow;
        const float bj = bias[col];
        #pragma unroll
        for (int r = 0; r < 8; ++r) {
            const int row = row0 + r + 8 * khalf;
            if (row < N) {
                float val = accs[jt][r] + bj;
                val = val > 0.f ? val : 0.f;
                out[row * D + col] = val + node_feats[row * D + col];
            }
        }
    }
}

// ---------------------------------------------------------------------------
extern "C" void kernel_launch(void* const* d_in, const int* in_sizes, int n_in,
                              void* d_out, int out_size, void* d_ws, size_t ws_size,
                              hipStream_t stream)
{
    const float* node_feats = (const float*)d_in[0];   // [N, 64]
    const float* edge_feats = (const float*)d_in[1];   // [E, 64]
    const int*   src        = (const int*)  d_in[2];   // [E]
    const int*   dst        = (const int*)  d_in[3];   // [E] (sorted)
    const float* W          = (const float*)d_in[4];   // [64, 64]
    const float* bias       = (const float*)d_in[5];   // [64]

    const int N = in_sizes[0] / D;
    const int E = in_sizes[2];

    float* agg = (float*)d_ws;                         // N*64*4 = 12.8 MB scratch
    float* out = (float*)d_out;

    // Kernel 1: one wave per node, 8 waves per 256-thread block.
    const int blocks1 = (N + 7) / 8;
    gcn_agg_kernel<<<blocks1, 256, 0, stream>>>(node_feats, edge_feats, src, dst,
                                                agg, N, E);

    // Kernel 2: 8 row-tiles (of 16 rows) per 256-thread block.
    const int tiles   = (N + 15) / 16;
    const int blocks2 = (tiles + 7) / 8;
    gcn_gemm_kernel<<<blocks2, 256, 0, stream>>>(agg, W, bias, node_feats, out, N);
}